// FQSM_56384330662191
// MI455X (gfx1250) — compile-verified
//
#include <hip/hip_runtime.h>
#include <math.h>

// ---------------- problem constants ----------------
#define BB      8
#define CCH     96
#define HH      224
#define WIN     7
#define NH      32          // HH/WIN
#define NWIN    1024        // NH*NH
#define PERK    24          // C/4
#define DSTATE  16
#define DTRANK  6
#define TOPK    256         // nearest square of 0.25*N
#define NSEL    16
#define LL      12544       // TOPK*49
#define CPROJ   38          // DTRANK + 2*DSTATE
#define LTILES  784         // LL/16

typedef float v2f __attribute__((ext_vector_type(2)));
typedef float v8f __attribute__((ext_vector_type(8)));

// ---------------- workspace layout (float elements) ----------------
static constexpr size_t OFF_PROBS  = 0;                            // B*N
static constexpr size_t OFF_RW     = OFF_PROBS  + (size_t)BB*NWIN; // B*TOPK
static constexpr size_t OFF_SEL    = OFF_RW     + (size_t)BB*TOPK; // B*TOPK (ints)
static constexpr size_t OFF_LOGITS = OFF_SEL    + (size_t)BB*TOPK;
static constexpr size_t OFF_XS     = OFF_LOGITS + (size_t)BB*NWIN;
static constexpr size_t OFF_DELTA  = OFF_XS     + (size_t)BB*4*PERK*LL;
static constexpr size_t OFF_BM     = OFF_DELTA  + (size_t)BB*4*PERK*LL;
static constexpr size_t OFF_CM     = OFF_BM     + (size_t)BB*4*LL*DSTATE;
static constexpr size_t OFF_Y      = OFF_CM     + (size_t)BB*4*LL*DSTATE;

// ============ K1: window mean-pool + router MLP -> logits ============
__global__ __launch_bounds__(128)
void router_kernel(const float* __restrict__ x,
                   const float* __restrict__ w1, const float* __restrict__ b1,
                   const float* __restrict__ w2, const float* __restrict__ b2,
                   float* __restrict__ logits) {
  __shared__ float pooled_s[CCH];
  __shared__ float hid_s[PERK];
  int bj = blockIdx.x;
  int b = bj / NWIN, j = bj % NWIN;
  int t = threadIdx.x;
  if (t < CCH) {
    const float* base = x + (((size_t)b*CCH + t)*HH + (size_t)(j >> 5)*WIN)*HH + (size_t)(j & 31)*WIN;
    float s = 0.f;
    #pragma unroll
    for (int wi = 0; wi < WIN; ++wi)
      #pragma unroll
      for (int wj = 0; wj < WIN; ++wj)
        s += base[wi*HH + wj];
    pooled_s[t] = s * (1.0f / 49.0f);
  }
  __syncthreads();
  if (t < PERK) {
    float a = b1[t];
    for (int c = 0; c < CCH; ++c) a += pooled_s[c] * w1[t*CCH + c];
    hid_s[t] = 0.5f * a * (1.0f + erff(a * 0.7071067811865475f));   // exact GELU
  }
  __syncthreads();
  if (t == 0) {
    float a = b2[0];
    #pragma unroll
    for (int i = 0; i < PERK; ++i) a += hid_s[i] * w2[i];
    logits[(size_t)b*NWIN + j] = a;
  }
}

// ============ K2: per-batch softmax + stable top-256 (rank by count) ============
__global__ __launch_bounds__(1024)
void topk_kernel(const float* __restrict__ logits,
                 float* __restrict__ probs, float* __restrict__ rw, int* __restrict__ sel) {
  __shared__ float sh[NWIN];
  int b = blockIdx.x, t = threadIdx.x;
  float lg = logits[(size_t)b*NWIN + t];
  sh[t] = lg; __syncthreads();
  for (int s = 512; s > 0; s >>= 1) { if (t < s) sh[t] = fmaxf(sh[t], sh[t+s]); __syncthreads(); }
  float mx = sh[0]; __syncthreads();
  float e = expf(lg - mx);
  sh[t] = e; __syncthreads();
  for (int s = 512; s > 0; s >>= 1) { if (t < s) sh[t] += sh[t+s]; __syncthreads(); }
  float inv = 1.0f / sh[0]; __syncthreads();
  float p = e * inv;
  probs[(size_t)b*NWIN + t] = p;
  sh[t] = p; __syncthreads();
  int rank = 0;
  for (int jj = 0; jj < NWIN; ++jj) {
    float pj = sh[jj];
    rank += (pj > p) || (pj == p && jj < t);      // matches jax.lax.top_k stable order
  }
  if (rank < TOPK) { sel[(size_t)b*TOPK + rank] = t; rw[(size_t)b*TOPK + rank] = p; }
}

// ============ K3: gather selected windows into 4 scan orders: xs[b,k,d,l] ============
__global__ __launch_bounds__(256)
void gather_kernel(const float* __restrict__ x, const int* __restrict__ sel,
                   float* __restrict__ xs) {
  size_t idx = (size_t)blockIdx.x*blockDim.x + threadIdx.x;
  if (idx >= (size_t)BB*4*PERK*LL) return;
  int l = (int)(idx % LL); size_t t = idx / LL;
  int d = (int)(t % PERK); t /= PERK;
  int k = (int)(t & 3);    int b = (int)(t >> 2);
  int lu = (k >= 2) ? (LL - 1 - l) : l;          // flipped directions
  int ni, nj, wi, wj;
  if ((k & 1) == 0) {                            // row-major scan
    wj = lu % 7; int t2 = lu / 7; wi = t2 % 7; t2 /= 7; nj = t2 & 15; ni = t2 >> 4;
  } else {                                       // column-major scan
    wi = lu % 7; int t2 = lu / 7; wj = t2 % 7; t2 /= 7; ni = t2 & 15; nj = t2 >> 4;
  }
  int w = sel[(size_t)b*TOPK + ni*NSEL + nj];
  int c = 4*d + k;                               // interleaved channel grouping (c%4==k)
  xs[idx] = x[(((size_t)b*CCH + c)*HH + (size_t)(w >> 5)*WIN + wi)*HH + (size_t)(w & 31)*WIN + wj];
}

// ============ K4: WMMA projections (x_dbl = Wp @ xs ; delta = softplus(Wdt @ dts + bias)) ============
// one wave per (b,k,l-tile of 16); V_WMMA_F32_16X16X4_F32 chains over K
__global__ __launch_bounds__(32)
void proj_kernel(const float* __restrict__ xs,
                 const float* __restrict__ xpw,      // (4,38,24)
                 const float* __restrict__ dtw_all,  // (4,24,6)
                 const float* __restrict__ dt_bias,  // (4,24)
                 float* __restrict__ delta, float* __restrict__ Bm, float* __restrict__ Cm) {
  __shared__ float lds[48*16];                   // x_dbl tile, rows padded 38->48
  int lane = threadIdx.x;
  int tile = blockIdx.x % LTILES;
  int t    = blockIdx.x / LTILES;
  int k = t & 3, b = t >> 2;
  int l0 = tile * 16;
  bool lo = lane < 16;
  int mrow = lane & 15;
  size_t bk = (size_t)(b*4 + k);
  const float* Wp    = xpw + (size_t)k*CPROJ*PERK;
  const float* xs_bk = xs + bk*PERK*LL;

  // A fragments (weights), ISA 16x4 f32 layout: lanes0-15 K0/K1, lanes16-31 K2/K3
  v2f afrag[3][6];
  #pragma unroll
  for (int m = 0; m < 3; ++m)
    #pragma unroll
    for (int q = 0; q < 6; ++q) {
      int c = 16*m + mrow;
      int col = 4*q + (lo ? 0 : 2);
      afrag[m][q].x = (c < CPROJ) ? Wp[c*PERK + col]     : 0.0f;
      afrag[m][q].y = (c < CPROJ) ? Wp[c*PERK + col + 1] : 0.0f;
    }

  v8f acc0 = {}, acc1 = {}, acc2 = {};
  #pragma unroll
  for (int q = 0; q < 6; ++q) {                  // K = 24 -> 6 chunks of 4
    int r0 = 4*q + (lo ? 0 : 2);
    v2f bfrag;
    bfrag.x = xs_bk[(size_t)r0*LL + l0 + mrow];
    bfrag.y = xs_bk[(size_t)(r0+1)*LL + l0 + mrow];
    acc0 = __builtin_amdgcn_wmma_f32_16x16x4_f32(false, afrag[0][q], false, bfrag, (short)0, acc0, false, false);
    acc1 = __builtin_amdgcn_wmma_f32_16x16x4_f32(false, afrag[1][q], false, bfrag, (short)0, acc1, false, false);
    acc2 = __builtin_amdgcn_wmma_f32_16x16x4_f32(false, afrag[2][q], false, bfrag, (short)0, acc2, false, false);
  }

  // spill x_dbl tile through LDS (C/D layout: VGPR r -> rows r / r+8)
  #pragma unroll
  for (int r = 0; r < 8; ++r) {
    int ro = r + (lo ? 0 : 8);
    lds[(ro     )*16 + mrow] = acc0[r];
    lds[(ro + 16)*16 + mrow] = acc1[r];
    lds[(ro + 32)*16 + mrow] = acc2[r];
  }
  __syncthreads();

  // emit B (rows 6..21) and C (rows 22..37) transposed to [l][n] for the scan
  for (int idx = lane; idx < 256; idx += 32) {
    int nst = idx & 15, col = idx >> 4;
    size_t o = (bk*LL + l0 + col)*DSTATE + nst;
    Bm[o] = lds[(DTRANK + nst)*16 + col];
    Cm[o] = lds[(DTRANK + DSTATE + nst)*16 + col];
  }

  // second WMMA chain: delta = softplus(Wdt(24x6) @ dts(6xL) + bias)
  const float* dtw = dtw_all + (size_t)k*PERK*DTRANK;
  #pragma unroll
  for (int m2 = 0; m2 < 2; ++m2) {
    v8f accd = {};
    #pragma unroll
    for (int q2 = 0; q2 < 2; ++q2) {             // K padded 6 -> 8
      int d2 = 16*m2 + mrow;
      int rr = 4*q2 + (lo ? 0 : 2);
      v2f a2;
      a2.x = (d2 < PERK && rr     < DTRANK) ? dtw[d2*DTRANK + rr]     : 0.0f;
      a2.y = (d2 < PERK && rr + 1 < DTRANK) ? dtw[d2*DTRANK + rr + 1] : 0.0f;
      v2f bf;                                    // garbage rows 6,7 get zero weight
      bf.x = lds[(rr    )*16 + mrow];
      bf.y = lds[(rr + 1)*16 + mrow];
      accd = __builtin_amdgcn_wmma_f32_16x16x4_f32(false, a2, false, bf, (short)0, accd, false, false);
    }
    #pragma unroll
    for (int r = 0; r < 8; ++r) {
      int drow = 16*m2 + r + (lo ? 0 : 8);
      if (drow < PERK) {
        float v = accd[r] + dt_bias[(size_t)k*PERK + drow];
        float sp = fmaxf(v, 0.0f) + log1pf(expf(-fabsf(v)));   // stable softplus
        delta[(bk*PERK + drow)*LL + l0 + mrow] = sp;
      }
    }
  }
}

// ============ K5: selective scan — 16 state-lanes per chain, 2 chains per wave ============
__global__ __launch_bounds__(256)
void scan_kernel(const float* __restrict__ delta, const float* __restrict__ xs,
                 const float* __restrict__ Bm, const float* __restrict__ Cm,
                 const float* __restrict__ A_logs, const float* __restrict__ Ds,
                 float* __restrict__ y) {
  int g = blockIdx.x * blockDim.x + threadIdx.x;
  int chain = g >> 4;
  int n = g & 15;
  if (chain >= BB*4*PERK) return;
  int d = chain % PERK; int t = chain / PERK;
  int k = t & 3, b = t >> 2;
  size_t bk = (size_t)(b*4 + k);
  const float* dp = delta + (bk*PERK + d)*LL;
  const float* up = xs    + (bk*PERK + d)*LL;
  const float* Bp = Bm + bk*LL*DSTATE + n;
  const float* Cp = Cm + bk*LL*DSTATE + n;
  float*       yp = y  + (bk*PERK + d)*LL;
  float A   = -expf(A_logs[(size_t)(k*PERK + d)*DSTATE + n]);
  float Dsv = Ds[(size_t)k*PERK + d];
  float h = 0.0f;
  for (int l = 0; l < LL; ++l) {
    __builtin_prefetch(Bp + (size_t)(l + 8)*DSTATE, 0, 0);
    __builtin_prefetch(Cp + (size_t)(l + 8)*DSTATE, 0, 0);
    float dv = dp[l];
    float uv = up[l];
    float Bt = Bp[(size_t)l*DSTATE];
    float Ct = Cp[(size_t)l*DSTATE];
    h = expf(dv * A) * h + (dv * uv) * Bt;
    float yl = h * Ct;
    yl += __shfl_xor(yl, 1, 32);
    yl += __shfl_xor(yl, 2, 32);
    yl += __shfl_xor(yl, 4, 32);
    yl += __shfl_xor(yl, 8, 32);                  // stays within each 16-lane half
    if (n == 0) yp[l] = yl + uv * Dsv;
  }
}

// ============ K6a: base residual for all windows: out = x*(1+prob) ============
__global__ __launch_bounds__(256)
void base_kernel(const float* __restrict__ x, const float* __restrict__ probs,
                 float* __restrict__ out) {
  size_t idx = (size_t)blockIdx.x*blockDim.x + threadIdx.x;
  if (idx >= (size_t)BB*NWIN*CCH*49) return;
  int p = (int)(idx % 49); size_t t = idx / 49;
  int c = (int)(t % CCH); t /= CCH;
  int j = (int)(t % NWIN); int b = (int)(t / NWIN);
  float base = x[(((size_t)b*CCH + c)*HH + (size_t)(j >> 5)*WIN + p/7)*HH + (size_t)(j & 31)*WIN + p%7];
  out[idx] = base * (1.0f + probs[(size_t)b*NWIN + j]);
}

// ============ K6b: overwrite selected windows: out = x + yout*rw (channel-permuted) ============
__global__ __launch_bounds__(256)
void sel_out_kernel(const float* __restrict__ x, const float* __restrict__ y,
                    const int* __restrict__ sel, const float* __restrict__ rw,
                    float* __restrict__ out) {
  size_t idx = (size_t)blockIdx.x*blockDim.x + threadIdx.x;
  if (idx >= (size_t)BB*CCH*LL) return;
  int lp = (int)(idx % LL); size_t t = idx / LL;
  int cc = (int)(t % CCH); int b = (int)(t / CCH);
  int k = cc / PERK, d = cc % PERK;
  // invert local_reverse: l' = ((ni*7+wi)*16 + nj)*7 + wj
  int wj = lp % 7; int t2 = lp / 7;
  int nj = t2 & 15; t2 >>= 4;
  int wi = t2 % 7;  int ni = t2 / 7;
  int lu = ((k & 1) == 0) ? (ni*784 + nj*49 + wi*7 + wj)
                          : (nj*784 + ni*49 + wj*7 + wi);
  int l  = (k >= 2) ? (LL - 1 - lu) : lu;
  float yv = y[(((size_t)b*4 + k)*PERK + d)*LL + l];
  int gslot = lp / 49, p = lp % 49;
  int w = sel[(size_t)b*TOPK + gslot];
  float base = x[(((size_t)b*CCH + cc)*HH + (size_t)(w >> 5)*WIN + p/7)*HH + (size_t)(w & 31)*WIN + p%7];
  out[(((size_t)b*NWIN + w)*CCH + cc)*49 + p] = base + yv * rw[(size_t)b*TOPK + gslot];
}

// ================================ launcher ================================
extern "C" void kernel_launch(void* const* d_in, const int* in_sizes, int n_in,
                              void* d_out, int out_size, void* d_ws, size_t ws_size,
                              hipStream_t stream) {
  (void)in_sizes; (void)n_in; (void)out_size; (void)ws_size;
  const float* x    = (const float*)d_in[0];
  const float* rw1  = (const float*)d_in[1];
  const float* rb1  = (const float*)d_in[2];
  const float* rw2  = (const float*)d_in[3];
  const float* rb2  = (const float*)d_in[4];
  const float* xpw  = (const float*)d_in[5];
  const float* dtw  = (const float*)d_in[6];
  const float* dtb  = (const float*)d_in[7];
  const float* Alg  = (const float*)d_in[8];
  const float* Ds   = (const float*)d_in[9];
  float* out = (float*)d_out;

  float* ws     = (float*)d_ws;
  float* probs  = ws + OFF_PROBS;
  float* rwv    = ws + OFF_RW;
  int*   sel    = (int*)(ws + OFF_SEL);
  float* logits = ws + OFF_LOGITS;
  float* xs     = ws + OFF_XS;
  float* delta  = ws + OFF_DELTA;
  float* Bm     = ws + OFF_BM;
  float* Cm     = ws + OFF_CM;
  float* yw     = ws + OFF_Y;

  router_kernel<<<BB*NWIN, 128, 0, stream>>>(x, rw1, rb1, rw2, rb2, logits);
  topk_kernel<<<BB, 1024, 0, stream>>>(logits, probs, rwv, sel);
  {
    size_t n = (size_t)BB*4*PERK*LL;
    gather_kernel<<<(unsigned)((n + 255)/256), 256, 0, stream>>>(x, sel, xs);
  }
  proj_kernel<<<BB*4*LTILES, 32, 0, stream>>>(xs, xpw, dtw, dtb, delta, Bm, Cm);
  scan_kernel<<<(BB*4*PERK*16 + 255)/256, 256, 0, stream>>>(delta, xs, Bm, Cm, Alg, Ds, yw);
  {
    size_t n = (size_t)BB*NWIN*CCH*49;
    base_kernel<<<(unsigned)((n + 255)/256), 256, 0, stream>>>(x, probs, out);
  }
  {
    size_t n = (size_t)BB*CCH*LL;
    sel_out_kernel<<<(unsigned)((n + 255)/256), 256, 0, stream>>>(x, yw, sel, rwv, out);
  }
}